// MultiHeadAttentionLayer_68461778698590
// MI455X (gfx1250) — compile-verified
//
#include <hip/hip_runtime.h>
#include <math.h>

// ---------------------------------------------------------------------------
// Graph-transformer attention layer for MI455X (gfx1250), wave32 + WMMA f32.
//   kernel1: Qh/Kh/Vh = LN(h@W + b)          (V_WMMA_F32_16X16X4_F32)
//   kernel2: proj_e/lpos via WMMA + LN, gather K/Q/V, clamp, per-head softmax,
//            e_out store, atomic scatter-sum of wV and z
//   kernel3: h_out = wV / (z + 1e-6)
//
// Weights are staged into LDS *pair-swizzled*: element n of K-rows (2p, 2p+1)
// stored adjacently, so every WMMA B fragment is a single ds_load_b64 into an
// even-aligned VGPR pair (no ds_load_2addr + v_mov packing).
// ---------------------------------------------------------------------------

typedef __attribute__((ext_vector_type(2))) float v2f;
typedef __attribute__((ext_vector_type(8))) float v8f;

#define LN_EPS 1e-5f

__device__ __forceinline__ v8f wmma4(v2f a, v2f b, v8f c) {
  // D = A(16x4) x B(4x16) + C(16x16), fp32
  return __builtin_amdgcn_wmma_f32_16x16x4_f32(
      /*neg_a=*/false, a, /*neg_b=*/false, b,
      /*c_mod=*/(short)0, c, /*reuse_a=*/false, /*reuse_b=*/false);
}

// Stage one 64x64 weight matrix into pair-swizzled LDS:
//   dstPairs[(k>>1)*64 + n][k&1] = W[k*64 + n]
__device__ __forceinline__ void stage_weights(v2f* dstPairs,
                                              const float* __restrict__ W) {
  float* d = (float*)dstPairs;
  for (int i = threadIdx.x; i < 4096; i += 256) {
    const int k = i >> 6;
    const int n = i & 63;
    d[(((k >> 1) * 64 + n) << 1) + (k & 1)] = W[i];
  }
}

// One 64-wide GEMM pass: C(16x64) = A(16x64) x W(64x64) using 4 column tiles.
// afrag: 16 K-step fragments.  Wp: pair-swizzled weights in LDS.
__device__ __forceinline__ void gemm64(const v2f afrag[16], const v2f* Wp,
                                       int m, int kh, v8f acc[4]) {
#pragma unroll
  for (int ct = 0; ct < 4; ++ct) {
    v8f c = {};
    const v2f* Wc = Wp + kh * 64 + ct * 16 + m;
#pragma unroll
    for (int kk = 0; kk < 16; ++kk) {
      // pair index p = 2*kk + kh  ->  K rows (4kk+2kh, 4kk+2kh+1)
      c = wmma4(afrag[kk], Wc[kk * 128], c);
    }
    acc[ct] = c;
  }
}

// Apply per-column bias, then LayerNorm each 64-wide row held in the WMMA
// C-layout (4 column tiles x v8f). Row r+8*kh lives in lanes [0..15] or
// [16..31] of accumulator element r; reduce across the 16-lane half.
__device__ __forceinline__ void bias_ln(const v8f acc[4], const float bias[4],
                                        const float g[4], const float b[4],
                                        float out[8][4]) {
#pragma unroll
  for (int r = 0; r < 8; ++r) {
    float x[4];
#pragma unroll
    for (int ct = 0; ct < 4; ++ct) x[ct] = acc[ct][r] + bias[ct];
    float ls = x[0] + x[1] + x[2] + x[3];
    float lq = x[0] * x[0] + x[1] * x[1] + x[2] * x[2] + x[3] * x[3];
#pragma unroll
    for (int off = 1; off < 16; off <<= 1) {
      ls += __shfl_xor(ls, off, 32);
      lq += __shfl_xor(lq, off, 32);
    }
    float mean = ls * (1.0f / 64.0f);
    float var  = lq * (1.0f / 64.0f) - mean * mean;  // biased variance
    float rstd = rsqrtf(var + LN_EPS);
#pragma unroll
    for (int ct = 0; ct < 4; ++ct)
      out[r][ct] = (x[ct] - mean) * rstd * g[ct] + b[ct];
  }
}

// ---------------------------------------------------------------------------
// Kernel 1: node projections Qh, Kh, Vh  (one 16-node tile per wave)
// ---------------------------------------------------------------------------
__global__ __launch_bounds__(256) void node_proj_kernel(
    const float* __restrict__ h,
    const float* __restrict__ Wq, const float* __restrict__ bq,
    const float* __restrict__ Wk, const float* __restrict__ bk,
    const float* __restrict__ Wv, const float* __restrict__ bv,
    const float* __restrict__ ln_g, const float* __restrict__ ln_b,
    float* __restrict__ Qh, float* __restrict__ Kh, float* __restrict__ Vh,
    int n_tiles) {
  __shared__ v2f Wl[3 * 2048];   // pair-swizzled Wq | Wk | Wv
  __shared__ float bl[3 * 64];
  __shared__ float gl[64], bbl[64];

  stage_weights(Wl, Wq);
  stage_weights(Wl + 2048, Wk);
  stage_weights(Wl + 4096, Wv);
  if (threadIdx.x < 64) {
    int i = threadIdx.x;
    bl[i]       = bq[i];
    bl[64 + i]  = bk[i];
    bl[128 + i] = bv[i];
    gl[i]  = ln_g[i];
    bbl[i] = ln_b[i];
  }
  __syncthreads();

  const int wid  = threadIdx.x >> 5;
  const int lane = threadIdx.x & 31;
  const int tile = blockIdx.x * 8 + wid;
  if (tile >= n_tiles) return;
  const int m  = lane & 15;   // column within tile / row M for A
  const int kh = lane >> 4;   // half selector

  // A fragments, shared across Q/K/V GEMMs. A(16x4) layout:
  // v0 = {K=0 | K=2}, v1 = {K=1 | K=3} in lane halves.
  const v2f* hrow = (const v2f*)(h + (size_t)(tile * 16 + m) * 64);
  v2f afrag[16];
#pragma unroll
  for (int kk = 0; kk < 16; ++kk) afrag[kk] = hrow[2 * kk + kh];

  float* outs[3] = {Qh, Kh, Vh};
  float gv[4], bv2[4];
#pragma unroll
  for (int ct = 0; ct < 4; ++ct) {
    gv[ct]  = gl[ct * 16 + m];
    bv2[ct] = bbl[ct * 16 + m];
  }

#pragma unroll
  for (int mat = 0; mat < 3; ++mat) {
    v8f acc[4];
    gemm64(afrag, Wl + mat * 2048, m, kh, acc);

    float bias[4];
#pragma unroll
    for (int ct = 0; ct < 4; ++ct) bias[ct] = bl[mat * 64 + ct * 16 + m];

    float y[8][4];
    bias_ln(acc, bias, gv, bv2, y);

    float* outp = outs[mat] + (size_t)tile * 16 * 64;
#pragma unroll
    for (int r = 0; r < 8; ++r) {
      const int row = r + 8 * kh;
      outp[row * 64 +  0 + m] = y[r][0];
      outp[row * 64 + 16 + m] = y[r][1];
      outp[row * 64 + 32 + m] = y[r][2];
      outp[row * 64 + 48 + m] = y[r][3];
    }
  }
}

// ---------------------------------------------------------------------------
// Kernel 2: fused edge kernel (one 16-edge tile per wave)
// ---------------------------------------------------------------------------
__global__ __launch_bounds__(256) void edge_kernel(
    const float* __restrict__ e, const float* __restrict__ sp,
    const int* __restrict__ src, const int* __restrict__ dst,
    const float* __restrict__ We, const float* __restrict__ be,
    const float* __restrict__ Wp, const float* __restrict__ bp,
    const float* __restrict__ ln_g, const float* __restrict__ ln_b,
    const float* __restrict__ pos_emb, const float* __restrict__ temperature,
    const float* __restrict__ Qh, const float* __restrict__ Kh,
    const float* __restrict__ Vh,
    float* __restrict__ e_out, float* __restrict__ wV, float* __restrict__ z,
    int n_tiles) {
  __shared__ v2f Wl[2 * 2048];   // pair-swizzled We | Wp
  __shared__ float bl[2 * 64];
  __shared__ float gl[64], bbl[64], pel[64];

  stage_weights(Wl, We);
  stage_weights(Wl + 2048, Wp);
  if (threadIdx.x < 64) {
    int i = threadIdx.x;
    bl[i]      = be[i];
    bl[64 + i] = bp[i];
    gl[i]  = ln_g[i];
    bbl[i] = ln_b[i];
    pel[i] = pos_emb[i];
  }
  __syncthreads();

  const int wid  = threadIdx.x >> 5;
  const int lane = threadIdx.x & 31;
  const int tile = blockIdx.x * 8 + wid;
  if (tile >= n_tiles) return;
  const int m  = lane & 15;
  const int kh = lane >> 4;

  const float invs = 1.0f / (sqrtf(8.0f) * temperature[0]);

  v8f accE[4], accP[4];
  {  // proj_e GEMM
    const v2f* arow = (const v2f*)(e + (size_t)(tile * 16 + m) * 64);
    v2f afrag[16];
#pragma unroll
    for (int kk = 0; kk < 16; ++kk) afrag[kk] = arow[2 * kk + kh];
    gemm64(afrag, Wl, m, kh, accE);
  }
  {  // lpos GEMM
    const v2f* arow = (const v2f*)(sp + (size_t)(tile * 16 + m) * 64);
    v2f afrag[16];
#pragma unroll
    for (int kk = 0; kk < 16; ++kk) afrag[kk] = arow[2 * kk + kh];
    gemm64(afrag, Wl + 2048, m, kh, accP);
  }

  float gv[4], bv2[4], biasE[4], biasP[4];
#pragma unroll
  for (int ct = 0; ct < 4; ++ct) {
    gv[ct]    = gl[ct * 16 + m];
    bv2[ct]   = bbl[ct * 16 + m];
    biasE[ct] = bl[ct * 16 + m];
    biasP[ct] = bl[64 + ct * 16 + m] + pel[ct * 16 + m];  // + pos_emb pre-LN
  }

  float pee[8][4], lpo[8][4];
  bias_ln(accE, biasE, gv, bv2, pee);
  bias_ln(accP, biasP, gv, bv2, lpo);

#pragma unroll
  for (int r = 0; r < 8; ++r) {
    const int erow = tile * 16 + r + 8 * kh;
    const int si = src[erow];
    const int di = dst[erow];
    const float* Kr = Kh + (size_t)si * 64;
    const float* Qr = Qh + (size_t)di * 64;
    const float* Vr = Vh + (size_t)si * 64;

    float sc[4];
#pragma unroll
    for (int ct = 0; ct < 4; ++ct) {
      const int n = ct * 16 + m;
      float s = Kr[n] * Qr[n] * invs + pee[r][ct] + lpo[r][ct];
      sc[ct] = fminf(fmaxf(s, -5.0f), 5.0f);
    }
    // softmax over each head (8 consecutive columns -> 8-lane groups)
#pragma unroll
    for (int ct = 0; ct < 4; ++ct) {
      float v  = sc[ct];
      float mx = v;
#pragma unroll
      for (int off = 1; off < 8; off <<= 1)
        mx = fmaxf(mx, __shfl_xor(mx, off, 32));
      float ex = __expf(v - mx);
      float s = ex;
#pragma unroll
      for (int off = 1; off < 8; off <<= 1) s += __shfl_xor(s, off, 32);
      sc[ct] = ex / s;
    }
    float* eo = e_out + (size_t)erow * 64;
#pragma unroll
    for (int ct = 0; ct < 4; ++ct) {
      const int n = ct * 16 + m;
      const float sv = sc[ct];
      eo[n] = sv;
      unsafeAtomicAdd(&wV[(size_t)di * 64 + n], Vr[n] * sv);
      unsafeAtomicAdd(&z[(size_t)di * 64 + n], sv);
    }
  }
}

// ---------------------------------------------------------------------------
// Kernel 3: h_out = wV / (z + 1e-6), in place over d_out's h_out region
// ---------------------------------------------------------------------------
__global__ __launch_bounds__(256) void finalize_kernel(float* __restrict__ hout,
                                                       const float* __restrict__ z,
                                                       int n) {
  int i = blockIdx.x * blockDim.x + threadIdx.x;
  if (i < n) hout[i] = hout[i] / (z[i] + 1e-6f);
}

// ---------------------------------------------------------------------------
extern "C" void kernel_launch(void* const* d_in, const int* in_sizes, int n_in,
                              void* d_out, int out_size, void* d_ws,
                              size_t ws_size, hipStream_t stream) {
  const float* h    = (const float*)d_in[0];
  const float* e    = (const float*)d_in[1];
  const float* sp   = (const float*)d_in[2];
  const int*   src  = (const int*)d_in[3];
  const int*   dst  = (const int*)d_in[4];
  const float* Wq   = (const float*)d_in[5];
  const float* bq   = (const float*)d_in[6];
  const float* Wk   = (const float*)d_in[7];
  const float* bk   = (const float*)d_in[8];
  const float* Wv   = (const float*)d_in[9];
  const float* bv   = (const float*)d_in[10];
  const float* We   = (const float*)d_in[11];
  const float* be   = (const float*)d_in[12];
  const float* Wp   = (const float*)d_in[13];
  const float* bp   = (const float*)d_in[14];
  const float* lg   = (const float*)d_in[15];
  const float* lb   = (const float*)d_in[16];
  const float* pe   = (const float*)d_in[17];
  const float* temp = (const float*)d_in[18];

  const int n_nodes = in_sizes[0] / 64;
  const int n_edges = in_sizes[1] / 64;

  // workspace: Qh | Kh | Vh | z   (each n_nodes*64 floats)
  float* Qh = (float*)d_ws;
  float* Kh = Qh + (size_t)n_nodes * 64;
  float* Vh = Kh + (size_t)n_nodes * 64;
  float* z  = Vh + (size_t)n_nodes * 64;

  float* hout = (float*)d_out;                       // wV accumulator, then h_out
  float* eout = hout + (size_t)n_nodes * 64;         // e_out

  hipMemsetAsync(hout, 0, (size_t)n_nodes * 64 * sizeof(float), stream);
  hipMemsetAsync(z,    0, (size_t)n_nodes * 64 * sizeof(float), stream);

  const int ntile_n = (n_nodes + 15) / 16;
  node_proj_kernel<<<(ntile_n + 7) / 8, 256, 0, stream>>>(
      h, Wq, bq, Wk, bk, Wv, bv, lg, lb, Qh, Kh, Vh, ntile_n);

  const int ntile_e = (n_edges + 15) / 16;
  edge_kernel<<<(ntile_e + 7) / 8, 256, 0, stream>>>(
      e, sp, src, dst, We, be, Wp, bp, lg, lb, pe, temp, Qh, Kh, Vh, eout,
      hout, z, ntile_e);

  const int n_h = n_nodes * 64;
  finalize_kernel<<<(n_h + 255) / 256, 256, 0, stream>>>(hout, z, n_h);
}